// GNN_62285615726745
// MI455X (gfx1250) — compile-verified
//
#include <hip/hip_runtime.h>
#include <math.h>

typedef __attribute__((ext_vector_type(2))) float v2f;
typedef __attribute__((ext_vector_type(4))) float f4;
typedef __attribute__((ext_vector_type(8))) float v8f;

#define FEAT 64
#define ROWS_PER_BLOCK 128
#define SA_STRIDE 68   // padded row stride for A tile: float4-aligned, conflict-free frag reads
#define SW_STRIDE 65   // padded row stride for transposed-W tile

// ---------------------------------------------------------------- zero
__global__ void GNN_zero_kernel(float* __restrict__ p, int n) {
    int i = blockIdx.x * blockDim.x + threadIdx.x;
    if (i < n) p[i] = 0.0f;
}

// ---------------------------------------------------------------- scatter-add (segment_sum of gathered rows)
// one lane per (edge, feature): agg[dst[e]][f] += x[src[e]][f]
__global__ void GNN_scatter_add_kernel(const float* __restrict__ x,
                                       const int* __restrict__ src,
                                       const int* __restrict__ dst,
                                       float* __restrict__ agg,
                                       int nEdges) {
    int i = blockIdx.x * blockDim.x + threadIdx.x;
    int e = i >> 6;
    int f = i & 63;
    if (e < nEdges) {
        int s = src[e];
        int d = dst[e];
        atomicAdd(&agg[d * FEAT + f], x[s * FEAT + f]);
    }
}

// ---------------------------------------------------------------- shared GEMM core helpers
// Each wave: 16 rows x 64 cols, acc[4] of v8f, K=64 in 16 steps of V_WMMA_F32_16X16X4_F32.

__device__ __forceinline__ void gnn_load_tiles(const float* __restrict__ A,
                                               const float* __restrict__ W,
                                               const float* __restrict__ bias,
                                               float* sA, float* sW, float* sb,
                                               int blockRow0, int nRows, int tid) {
    // W transposed into LDS: sW[k*SW_STRIDE + h] = W[h*64 + k]
    for (int idx = tid; idx < FEAT * FEAT; idx += 256) {
        int h = idx >> 6;
        int k = idx & 63;
        sW[k * SW_STRIDE + h] = W[idx];
    }
    if (tid < FEAT) sb[tid] = bias[tid];
    // A rows, coalesced float4, padded LDS rows
#pragma unroll
    for (int i = 0; i < (ROWS_PER_BLOCK * FEAT / 4) / 256; ++i) {   // 8 iters
        int f4i = tid + i * 256;
        int fl  = f4i * 4;
        int r   = fl >> 6;
        int c   = fl & 63;
        f4 v = {0.f, 0.f, 0.f, 0.f};
        int grow = blockRow0 + r;
        if (grow < nRows) v = *(const f4*)&A[grow * FEAT + c];
        *(f4*)&sA[r * SA_STRIDE + c] = v;
    }
}

#define GNN_WMMA_BODY(acc)                                                         \
    _Pragma("unroll")                                                              \
    for (int kt = 0; kt < 16; ++kt) {                                              \
        int kb = kt * 4 + 2 * hi;                                                  \
        const float* ap = &sA[(w * 16 + m) * SA_STRIDE + kb];                      \
        v2f a = { ap[0], ap[1] };                                                  \
        const float* bp0 = &sW[kb * SW_STRIDE];                                    \
        const float* bp1 = &sW[(kb + 1) * SW_STRIDE];                              \
        _Pragma("unroll")                                                          \
        for (int nt = 0; nt < 4; ++nt) {                                           \
            v2f b = { bp0[nt * 16 + m], bp1[nt * 16 + m] };                        \
            acc[nt] = __builtin_amdgcn_wmma_f32_16x16x4_f32(                       \
                false, a, false, b, (short)0, acc[nt], false, false);              \
        }                                                                          \
    }

// ---------------------------------------------------------------- GEMM1: h = relu(agg @ W1^T + b1)
__global__ __launch_bounds__(256) void GNN_gemm_bias_relu_kernel(
        const float* __restrict__ A, const float* __restrict__ W,
        const float* __restrict__ bias, float* __restrict__ Hout, int nRows) {
    __shared__ float sA[ROWS_PER_BLOCK * SA_STRIDE];
    __shared__ float sW[FEAT * SW_STRIDE];
    __shared__ float sb[FEAT];
    int tid = threadIdx.x;
    int blockRow0 = blockIdx.x * ROWS_PER_BLOCK;
    gnn_load_tiles(A, W, bias, sA, sW, sb, blockRow0, nRows, tid);
    __syncthreads();

    int lane = tid & 31, w = tid >> 5;
    int m = lane & 15, hi = lane >> 4;
    int waveRow0 = blockRow0 + w * 16;
    if (waveRow0 < nRows) {                       // wave-uniform guard; EXEC all-ones for WMMA
        v8f acc[4] = {};
        GNN_WMMA_BODY(acc)
#pragma unroll
        for (int nt = 0; nt < 4; ++nt) {
            float bv = sb[nt * 16 + m];
#pragma unroll
            for (int r = 0; r < 8; ++r) {
                int row = waveRow0 + r + 8 * hi;  // C layout: M = r + 8*hi, N = m
                float v = acc[nt][r] + bv;
                v = v > 0.f ? v : 0.f;
                Hout[row * FEAT + nt * 16 + m] = v;
            }
        }
    }
}

// ---------------------------------------------------------------- GEMM2 fused with h2^T @ W3^T reduction
// classAcc[c] += sum_n (agg2 @ W2^T + b2)[n][c] * W3[n]
__global__ __launch_bounds__(256) void GNN_gemm_reduce_kernel(
        const float* __restrict__ A, const float* __restrict__ W,
        const float* __restrict__ bias, const float* __restrict__ W3,
        float* __restrict__ classAcc, int nRows) {
    __shared__ float sA[ROWS_PER_BLOCK * SA_STRIDE];
    __shared__ float sW[FEAT * SW_STRIDE];
    __shared__ float sb[FEAT];
    __shared__ float sAcc[FEAT];
    int tid = threadIdx.x;
    int blockRow0 = blockIdx.x * ROWS_PER_BLOCK;
    if (tid < FEAT) sAcc[tid] = 0.f;
    gnn_load_tiles(A, W, bias, sA, sW, sb, blockRow0, nRows, tid);
    __syncthreads();

    int lane = tid & 31, w = tid >> 5;
    int m = lane & 15, hi = lane >> 4;
    int waveRow0 = blockRow0 + w * 16;
    if (waveRow0 < nRows) {
        v8f acc[4] = {};
        GNN_WMMA_BODY(acc)
#pragma unroll
        for (int nt = 0; nt < 4; ++nt) {
            float bv = sb[nt * 16 + m];
            float partial = 0.f;
#pragma unroll
            for (int r = 0; r < 8; ++r) {
                int row = waveRow0 + r + 8 * hi;
                float v = acc[nt][r] + bv;      // h2[row][c], c = nt*16 + m
                partial += v * W3[row];
            }
            atomicAdd(&sAcc[nt * 16 + m], partial);  // LDS float atomic (ds_add_f32)
        }
    }
    __syncthreads();
    if (tid < FEAT) atomicAdd(&classAcc[tid], sAcc[tid]);
}

// ---------------------------------------------------------------- finalize: sigmoid(classAcc + b3)
__global__ void GNN_finalize_kernel(const float* __restrict__ classAcc,
                                    const float* __restrict__ b3,
                                    float* __restrict__ out) {
    int c = threadIdx.x;
    if (c < FEAT) {
        float v = classAcc[c] + b3[0];
        out[c] = 1.0f / (1.0f + expf(-v));
    }
}

// ---------------------------------------------------------------- launcher
extern "C" void kernel_launch(void* const* d_in, const int* in_sizes, int n_in,
                              void* d_out, int out_size, void* d_ws, size_t ws_size,
                              hipStream_t stream) {
    const float* inputs = (const float*)d_in[0];
    const float* W1     = (const float*)d_in[1];
    const float* b1     = (const float*)d_in[2];
    const float* W2     = (const float*)d_in[3];
    const float* b2     = (const float*)d_in[4];
    const float* W3     = (const float*)d_in[5];
    const float* b3     = (const float*)d_in[6];
    const int*   src    = (const int*)d_in[7];
    const int*   dst    = (const int*)d_in[8];
    float*       out    = (float*)d_out;

    int nNodes = in_sizes[0] / FEAT;
    int nEdges = in_sizes[7];

    float* wsf      = (float*)d_ws;
    float* classAcc = wsf;                       // 64 floats
    float* agg      = wsf + 64;                  // nNodes*64
    float* hbuf     = agg + (size_t)nNodes * FEAT;

    int nAgg = nNodes * FEAT;
    int nz   = 64 + nAgg;
    int workE = nEdges * FEAT;
    int gemmBlocks = (nNodes + ROWS_PER_BLOCK - 1) / ROWS_PER_BLOCK;

    // 1) zero classAcc + agg
    GNN_zero_kernel<<<(nz + 255) / 256, 256, 0, stream>>>(wsf, nz);
    // 2) agg = segment_sum(inputs[src], dst)
    GNN_scatter_add_kernel<<<(workE + 255) / 256, 256, 0, stream>>>(inputs, src, dst, agg, nEdges);
    // 3) h = relu(agg @ W1^T + b1)
    GNN_gemm_bias_relu_kernel<<<gemmBlocks, 256, 0, stream>>>(agg, W1, b1, hbuf, nNodes);
    // 4) zero agg
    GNN_zero_kernel<<<(nAgg + 255) / 256, 256, 0, stream>>>(agg, nAgg);
    // 5) agg = segment_sum(h[src], dst)
    GNN_scatter_add_kernel<<<(workE + 255) / 256, 256, 0, stream>>>(hbuf, src, dst, agg, nEdges);
    // 6) classAcc = sum_n (agg @ W2^T + b2)[n][:] * W3[n]   (fused GEMM + reduction)
    GNN_gemm_reduce_kernel<<<gemmBlocks, 256, 0, stream>>>(agg, W2, b2, W3, classAcc, nNodes);
    // 7) out = sigmoid(classAcc + b3)
    GNN_finalize_kernel<<<1, 64, 0, stream>>>(classAcc, b3, out);
}